// Bahadanau_Attention_29523605192707
// MI455X (gfx1250) — compile-verified
//
#include <hip/hip_runtime.h>
#include <hip/hip_bf16.h>

typedef __attribute__((ext_vector_type(2))) float v2f;
typedef __attribute__((ext_vector_type(8))) float v8f;

#define B_    32
#define L_    2048
#define V_    512
#define H_    1024
#define D_    512
#define NNZ_  512
#define NCHUNK 4

// workspace byte offsets (all 256B aligned)
#define OFF_MASK    0          // B*L u32          = 262144
#define OFF_COUNT   262144     // B u32 (pad 1KB)
#define OFF_LIST    263168     // B*NNZ u32        = 65536
#define OFF_SCORE   328704     // B*NNZ f32        = 65536
#define OFF_ALPHAC  394240     // B*NNZ f32        = 65536
#define OFF_U       459776     // H f32            = 4096
#define OFF_S2      463872     // B f32 (pad 1KB)
#define OFF_PART    464896     // NCHUNK*B*D f32   = 262144  (end 727040)

// ---------- pass 0: zero mask/counts and alpha output region ----------
__global__ void k_init(unsigned* __restrict__ mask, unsigned* __restrict__ count,
                       float* __restrict__ alpha_out) {
    int i = blockIdx.x * blockDim.x + threadIdx.x;
    if (i < B_ * L_) { mask[i] = 0u; alpha_out[i] = 0.f; }
    if (i < B_) count[i] = 0u;
}

// ---------- pass 1: scatter sparse indices into dense mask (dup-safe) ----------
__global__ void k_mask(const int* __restrict__ sidx, unsigned* __restrict__ mask,
                       int nnz_total) {
    int i = blockIdx.x * blockDim.x + threadIdx.x;
    if (i >= nnz_total) return;
    int b = sidx[i];
    int l = sidx[nnz_total + i];
    if ((unsigned)b < B_ && (unsigned)l < L_) mask[(size_t)b * L_ + l] = 1u;
}

// ---------- pass 2: deterministic ordered compaction (ascending l) ----------
__global__ void k_compact(const unsigned* __restrict__ mask,
                          unsigned* __restrict__ list, unsigned* __restrict__ count) {
    __shared__ unsigned sc[256];
    const int b = blockIdx.x;
    const int t = threadIdx.x;
    const int per = L_ / 256;          // 8
    unsigned local[8]; int nl = 0;
    for (int k = 0; k < per; ++k) {
        unsigned l = (unsigned)(t * per + k);
        if (mask[(size_t)b * L_ + l]) local[nl++] = l;
    }
    sc[t] = (unsigned)nl;
    __syncthreads();
    for (int off = 1; off < 256; off <<= 1) {     // Hillis-Steele inclusive scan
        unsigned v = (t >= off) ? sc[t - off] : 0u;
        __syncthreads();
        sc[t] += v;
        __syncthreads();
    }
    unsigned base = b * NNZ_ + sc[t] - (unsigned)nl;  // exclusive prefix
    for (int k = 0; k < nl; ++k) list[base + k] = local[k];
    if (t == 255) count[b] = sc[255];
}

// ---------- pass 3: u[h] = sum_v W1[v,h]*W2[v]  (V x H GEMV) ----------
__global__ void k_u(const float* __restrict__ W1, const float* __restrict__ W2,
                    float* __restrict__ u) {
    int h = blockIdx.x * blockDim.x + threadIdx.x;
    if (h >= H_) return;
    float s = 0.f;
    #pragma unroll 4
    for (int v = 0; v < V_; ++v) s += W1[(size_t)v * H_ + h] * W2[v];
    u[h] = s;
}

// ---------- pass 4: s2[b] = factor * (dhs[b,:] . u) ----------
__global__ void k_s2(const float* __restrict__ dhs, const float* __restrict__ u,
                     const int* __restrict__ sv, float* __restrict__ s2) {
    __shared__ float red[256];
    int b = blockIdx.x, t = threadIdx.x;
    float s = 0.f;
    for (int h = t; h < H_; h += 256) s += dhs[(size_t)b * H_ + h] * u[h];
    red[t] = s; __syncthreads();
    for (int off = 128; off > 0; off >>= 1) {
        if (t < off) red[t] += red[t + off];
        __syncthreads();
    }
    if (t == 0) { float f = (sv[0] == 1) ? 1.f : 2.f; s2[b] = f * red[0]; }
}

// ---------- pass 5: scores via V_WMMA_F32_16X16X4_F32 ----------
// A: 16 gathered bdi rows x 4 K (lanes 0-15 hold K={0,1}, lanes 16-31 K={2,3})
// B: W2[k] broadcast across all 16 columns (same K split); C columns all equal.
__global__ __launch_bounds__(32) void k_scores(const float* __restrict__ bdi,
                                               const float* __restrict__ W2,
                                               const unsigned* __restrict__ list,
                                               const unsigned* __restrict__ count,
                                               const float* __restrict__ s2,
                                               float* __restrict__ scores) {
    const int b = blockIdx.y;
    const int t = blockIdx.x;                 // 16-row tile index
    const int cnt = (int)count[b];
    if (t * 16 >= cnt) return;                // uniform over the wave
    const int lane = threadIdx.x;             // wave32
    const int m    = lane & 15;               // A-matrix row
    const int half = lane >> 4;               // K pair selector
    const int j = t * 16 + m;
    const bool valid = j < cnt;
    const unsigned row = valid ? list[b * NNZ_ + j] : 0u;
    const float* abase = bdi + ((size_t)b * L_ + row) * V_ + 2 * half;
    const float* wbase = W2 + 2 * half;
    v8f acc = {};
    #pragma unroll 4
    for (int k = 0; k < V_; k += 4) {
        v2f a = *(const v2f*)(abase + k);
        if (!valid) { a.x = 0.f; a.y = 0.f; }   // v_cndmask, EXEC stays full
        v2f w = *(const v2f*)(wbase + k);
        acc = __builtin_amdgcn_wmma_f32_16x16x4_f32(
            /*neg_a=*/false, a, /*neg_b=*/false, w,
            /*c_mod=*/(short)0, acc, /*reuse_a=*/false, /*reuse_b=*/false);
    }
    const float sb = s2[b];
    // C layout: VGPR r -> M=r (lanes 0-15), M=r+8 (lanes 16-31); all N equal.
    if (lane == 0) {
        #pragma unroll
        for (int r = 0; r < 8; ++r) {
            int jj = t * 16 + r;
            if (jj < cnt) scores[b * NNZ_ + jj] = acc[r] + sb;
        }
    } else if (lane == 16) {
        #pragma unroll
        for (int r = 0; r < 8; ++r) {
            int jj = t * 16 + 8 + r;
            if (jj < cnt) scores[b * NNZ_ + jj] = acc[r] + sb;
        }
    }
}

// ---------- pass 6: masked softmax per batch, scatter alpha ----------
__global__ void k_softmax(const float* __restrict__ scores,
                          const unsigned* __restrict__ list,
                          const unsigned* __restrict__ count,
                          float* __restrict__ alpha_c, float* __restrict__ alpha_out) {
    __shared__ float red[256];
    int b = blockIdx.x, t = threadIdx.x;
    int cnt = (int)count[b];
    float m = -3.402823466e+38f;
    for (int j = t; j < cnt; j += 256) m = fmaxf(m, scores[b * NNZ_ + j]);
    red[t] = m; __syncthreads();
    for (int off = 128; off > 0; off >>= 1) {
        if (t < off) red[t] = fmaxf(red[t], red[t + off]);
        __syncthreads();
    }
    float smax = red[0]; __syncthreads();
    float s = 0.f;
    for (int j = t; j < cnt; j += 256) {
        float e = __expf(scores[b * NNZ_ + j] - smax);
        alpha_c[b * NNZ_ + j] = e;
        s += e;
    }
    red[t] = s; __syncthreads();
    for (int off = 128; off > 0; off >>= 1) {
        if (t < off) red[t] += red[t + off];
        __syncthreads();
    }
    float inv = 1.f / red[0];
    for (int j = t; j < cnt; j += 256) {
        float a = alpha_c[b * NNZ_ + j] * inv;
        alpha_c[b * NNZ_ + j] = a;
        alpha_out[(size_t)b * L_ + list[b * NNZ_ + j]] = a;
    }
}

// ---------- pass 7: partial weighted sums over fixed j-chunks ----------
__global__ __launch_bounds__(512) void k_part(const float* __restrict__ att,
                                              const unsigned* __restrict__ list,
                                              const float* __restrict__ alpha_c,
                                              const unsigned* __restrict__ count,
                                              float* __restrict__ part) {
    __shared__ unsigned ll[(NNZ_ + NCHUNK - 1) / NCHUNK];
    __shared__ float    la[(NNZ_ + NCHUNK - 1) / NCHUNK];
    int b = blockIdx.y, c = blockIdx.x, t = threadIdx.x;
    int cnt  = (int)count[b];
    int clen = (cnt + NCHUNK - 1) / NCHUNK;
    int j0 = c * clen;
    int j1 = min(cnt, j0 + clen);
    int n = j1 - j0;
    if (t < n) { ll[t] = list[b * NNZ_ + j0 + t]; la[t] = alpha_c[b * NNZ_ + j0 + t]; }
    __syncthreads();
    float acc = 0.f;
    #pragma unroll 2
    for (int k = 0; k < n; ++k)
        acc += la[k] * att[((size_t)b * L_ + ll[k]) * D_ + t];
    part[((size_t)c * B_ + b) * D_ + t] = acc;
}

// ---------- pass 8: fixed-order chunk reduction -> c[b,d] ----------
__global__ void k_csum(const float* __restrict__ part, float* __restrict__ cout) {
    int i = blockIdx.x * blockDim.x + threadIdx.x;
    if (i >= B_ * D_) return;
    int b = i / D_, d = i % D_;
    float s = 0.f;
    #pragma unroll
    for (int c = 0; c < NCHUNK; ++c) s += part[((size_t)c * B_ + b) * D_ + d];
    cout[i] = s;
}

extern "C" void kernel_launch(void* const* d_in, const int* in_sizes, int n_in,
                              void* d_out, int out_size, void* d_ws, size_t ws_size,
                              hipStream_t stream) {
    const float* dhs = (const float*)d_in[0];   // [B,H]
    const float* bdi = (const float*)d_in[1];   // [B,L,V]
    const float* att = (const float*)d_in[2];   // [B,L,D]
    const int*   sidx= (const int*)d_in[3];     // [2, B*NNZ]
    const int*   sv  = (const int*)d_in[4];     // scalar
    const float* W1  = (const float*)d_in[5];   // [V,H]
    const float* W2  = (const float*)d_in[6];   // [V,1]
    float* out = (float*)d_out;
    float* c_out     = out;                     // [B,D]
    float* alpha_out = out + B_ * D_;           // [B,L]

    char* ws = (char*)d_ws;
    unsigned* mask   = (unsigned*)(ws + OFF_MASK);
    unsigned* count  = (unsigned*)(ws + OFF_COUNT);
    unsigned* list   = (unsigned*)(ws + OFF_LIST);
    float* scores    = (float*)(ws + OFF_SCORE);
    float* alpha_c   = (float*)(ws + OFF_ALPHAC);
    float* u         = (float*)(ws + OFF_U);
    float* s2        = (float*)(ws + OFF_S2);
    float* part      = (float*)(ws + OFF_PART);

    int nnz_total = in_sizes[3] / 2;

    k_init   <<<(B_ * L_ + 255) / 256, 256, 0, stream>>>(mask, count, alpha_out);
    k_mask   <<<(nnz_total + 255) / 256, 256, 0, stream>>>(sidx, mask, nnz_total);
    k_compact<<<B_, 256, 0, stream>>>(mask, list, count);
    k_u      <<<(H_ + 255) / 256, 256, 0, stream>>>(W1, W2, u);
    k_s2     <<<B_, 256, 0, stream>>>(dhs, u, sv, s2);
    dim3 gs(NNZ_ / 16, B_);
    k_scores <<<gs, 32, 0, stream>>>(bdi, W2, list, count, s2, scores);
    k_softmax<<<B_, 256, 0, stream>>>(scores, list, count, alpha_c, alpha_out);
    dim3 gp(NCHUNK, B_);
    k_part   <<<gp, D_, 0, stream>>>(att, list, alpha_c, count, part);
    k_csum   <<<(B_ * D_ + 255) / 256, 256, 0, stream>>>(part, c_out);
}